// ParticleGNN_83820581749133
// MI455X (gfx1250) — compile-verified
//
#include <hip/hip_runtime.h>
#include <math.h>

#define HH 256
#define NHEADS 4
#define CH 64

typedef float v2f __attribute__((ext_vector_type(2)));
typedef float v8f __attribute__((ext_vector_type(8)));

__device__ __forceinline__ float gelu_exact(float v) {
    return 0.5f * v * (1.0f + erff(v * 0.70710678118654752f));
}

// ---------------- encoder: h = gelu(x @ enc_w + enc_b), K=7 ----------------
__global__ void encoder_kernel(const float* __restrict__ x, const float* __restrict__ w,
                               const float* __restrict__ b, float* __restrict__ out, int n) {
    int idx = blockIdx.x * blockDim.x + threadIdx.x;
    if (idx >= n * HH) return;
    int node = idx >> 8;
    int f = idx & 255;
    const float* xr = x + (long)node * 7;
    float acc = b[f];
#pragma unroll
    for (int j = 0; j < 7; ++j) acc += xr[j] * w[j * HH + f];
    out[idx] = gelu_exact(acc);
}

// ---------------- WMMA fp32 GEMM: C[M,NCols] (+)= A[M,K] @ B[K,NCols] ------
// block = 256 threads = 8 waves (wave32); block tile 32x64; wave tile 16x16.
// ACT: 0=none 1=relu 2=gelu
template <int ACT, bool ACCUM, bool BIAS>
__global__ void __launch_bounds__(256)
wmma_gemm_kernel(const float* __restrict__ A, const float* __restrict__ B,
                 const float* __restrict__ bias, float* __restrict__ Cmat,
                 int M, int Kdim, int NCols) {
    const int KC = 128;
    __shared__ float As[32][KC + 4];
    __shared__ float Bs[KC][64 + 4];

    int tid = threadIdx.x;
    int wave = tid >> 5;
    int lane = tid & 31;
    int waveM = wave >> 2;  // 0..1
    int waveN = wave & 3;   // 0..3
    int rowBlk = blockIdx.y * 32;
    int colBlk = blockIdx.x * 64;

    v8f acc = {};
    int ml = lane & 15;
    int kh = (lane >> 4) * 2;  // lanes 0-15 -> K0/K1, lanes 16-31 -> K2/K3 (ISA layout)

    for (int k0 = 0; k0 < Kdim; k0 += KC) {
        for (int i = tid; i < 32 * KC; i += 256) {
            int r = i / KC, c = i % KC;
            As[r][c] = A[(long)(rowBlk + r) * Kdim + (k0 + c)];
        }
        for (int i = tid; i < KC * 64; i += 256) {
            int r = i >> 6, c = i & 63;
            Bs[r][c] = B[(long)(k0 + r) * NCols + (colBlk + c)];
        }
        __syncthreads();
#pragma unroll
        for (int kk = 0; kk < KC / 4; ++kk) {
            v2f a, b;
            a.x = As[waveM * 16 + ml][4 * kk + kh];
            a.y = As[waveM * 16 + ml][4 * kk + kh + 1];
            b.x = Bs[4 * kk + kh][waveN * 16 + ml];
            b.y = Bs[4 * kk + kh + 1][waveN * 16 + ml];
            acc = __builtin_amdgcn_wmma_f32_16x16x4_f32(false, a, false, b,
                                                        (short)0, acc, false, false);
        }
        __syncthreads();
    }

    // D layout: VGPR i -> row (i) for lanes 0-15, row (8+i) for lanes 16-31; N = lane&15
    int col = colBlk + waveN * 16 + ml;
    int rbase = rowBlk + waveM * 16 + ((lane >> 4) ? 8 : 0);
#pragma unroll
    for (int i = 0; i < 8; ++i) {
        long off = (long)(rbase + i) * NCols + col;
        float v = acc[i];
        if (ACCUM) v += Cmat[off];
        if (BIAS) v += bias[col];
        if (ACT == 1) v = fmaxf(v, 0.0f);
        else if (ACT == 2) v = gelu_exact(v);
        Cmat[off] = v;
    }
}

// ---------------- GAT attention scores per (node, head) --------------------
__global__ void att_score_kernel(const float* __restrict__ hh, const float* __restrict__ att_src,
                                 const float* __restrict__ att_dst, float* __restrict__ asrc,
                                 float* __restrict__ adst, int n) {
    int idx = blockIdx.x * blockDim.x + threadIdx.x;
    if (idx >= n * NHEADS) return;
    int node = idx >> 2;
    int h = idx & 3;
    const float* hp = hh + (long)node * HH + h * CH;
    float s = 0.f, d = 0.f;
#pragma unroll 8
    for (int c = 0; c < CH; ++c) {
        float v = hp[c];
        s += v * att_src[h * CH + c];
        d += v * att_dst[h * CH + c];
    }
    asrc[idx] = s;
    adst[idx] = d;
}

// ---------------- CSR build (by destination = col) -------------------------
__global__ void deg_kernel(const int* __restrict__ col, int* __restrict__ deg, int e) {
    int i = blockIdx.x * blockDim.x + threadIdx.x;
    if (i < e) atomicAdd(&deg[col[i]], 1);
}

__global__ void scan_kernel(const int* __restrict__ deg, int* __restrict__ rowptr, int n, int e) {
    __shared__ int tsum[256];
    int t = threadIdx.x;
    int per = (n + 255) / 256;
    int s0 = t * per;
    int s1 = s0 + per; if (s1 > n) s1 = n;
    int local = 0;
    for (int i = s0; i < s1; ++i) local += deg[i];
    tsum[t] = local;
    __syncthreads();
    for (int off = 1; off < 256; off <<= 1) {
        int v = (t >= off) ? tsum[t - off] : 0;
        __syncthreads();
        tsum[t] += v;
        __syncthreads();
    }
    int run = (t == 0) ? 0 : tsum[t - 1];
    for (int i = s0; i < s1; ++i) { rowptr[i] = run; run += deg[i]; }
    if (t == 0) rowptr[n] = e;
}

__global__ void fill_csr_kernel(const int* __restrict__ row, const int* __restrict__ col,
                                const int* __restrict__ rowptr, int* __restrict__ fillpos,
                                int* __restrict__ csr_src, int e) {
    int i = blockIdx.x * blockDim.x + threadIdx.x;
    if (i >= e) return;
    int c = col[i];
    int p = atomicAdd(&fillpos[c], 1);
    csr_src[rowptr[c] + p] = row[i];
}

__global__ void dinv_kernel(const int* __restrict__ deg, float* __restrict__ dinv, int n) {
    int i = blockIdx.x * blockDim.x + threadIdx.x;
    if (i >= n) return;
    int d = deg[i];
    dinv[i] = (d > 0) ? rsqrtf((float)d) : 0.0f;
}

// ---------------- GAT softmax + aggregate: one wave per destination node ----
__global__ void __launch_bounds__(256)
gat_aggregate_kernel(const float* __restrict__ hh, const float* __restrict__ asrc,
                     const float* __restrict__ adst, const int* __restrict__ rowptr,
                     const int* __restrict__ csr_src, const float* __restrict__ gat_b,
                     float* __restrict__ out, int n) {
    int wid = blockIdx.x * (blockDim.x >> 5) + (threadIdx.x >> 5);
    if (wid >= n) return;
    int lane = threadIdx.x & 31;
    int node = wid;
    int start = rowptr[node], end = rowptr[node + 1];

    float ad[NHEADS], m[NHEADS], den[NHEADS];
#pragma unroll
    for (int h = 0; h < NHEADS; ++h) ad[h] = adst[node * NHEADS + h];

    // pass 1: segment max (self-loop included; identical on all lanes)
#pragma unroll
    for (int h = 0; h < NHEADS; ++h) {
        float s = asrc[node * NHEADS + h] + ad[h];
        m[h] = (s > 0.f) ? s : 0.2f * s;
    }
    for (int i = start + lane; i < end; i += 32) {
        int src = csr_src[i];
#pragma unroll
        for (int h = 0; h < NHEADS; ++h) {
            float s = asrc[src * NHEADS + h] + ad[h];
            s = (s > 0.f) ? s : 0.2f * s;
            m[h] = fmaxf(m[h], s);
        }
    }
#pragma unroll
    for (int h = 0; h < NHEADS; ++h)
        for (int off = 16; off > 0; off >>= 1)
            m[h] = fmaxf(m[h], __shfl_xor(m[h], off, 32));

    // pass 2: denominator (self-loop counted on lane 0 only)
#pragma unroll
    for (int h = 0; h < NHEADS; ++h) {
        float s = asrc[node * NHEADS + h] + ad[h];
        s = (s > 0.f) ? s : 0.2f * s;
        den[h] = (lane == 0) ? expf(s - m[h]) : 0.f;
    }
    for (int i = start + lane; i < end; i += 32) {
        int src = csr_src[i];
#pragma unroll
        for (int h = 0; h < NHEADS; ++h) {
            float s = asrc[src * NHEADS + h] + ad[h];
            s = (s > 0.f) ? s : 0.2f * s;
            den[h] += expf(s - m[h]);
        }
    }
#pragma unroll
    for (int h = 0; h < NHEADS; ++h)
        for (int off = 16; off > 0; off >>= 1)
            den[h] += __shfl_xor(den[h], off, 32);

    // pass 3: aggregate (lanes parallel over channels; edges serial)
    float acc[8];
    {
        float alpha[NHEADS];
#pragma unroll
        for (int h = 0; h < NHEADS; ++h) {
            float s = asrc[node * NHEADS + h] + ad[h];
            s = (s > 0.f) ? s : 0.2f * s;
            alpha[h] = expf(s - m[h]) / den[h];
        }
#pragma unroll
        for (int j = 0; j < 8; ++j) {
            int c = j * 32 + lane;
            acc[j] = alpha[c >> 6] * hh[(long)node * HH + c];
        }
    }
    for (int i = start; i < end; ++i) {
        int src = csr_src[i];
        float alpha[NHEADS];
#pragma unroll
        for (int h = 0; h < NHEADS; ++h) {
            float s = asrc[src * NHEADS + h] + ad[h];
            s = (s > 0.f) ? s : 0.2f * s;
            alpha[h] = expf(s - m[h]) / den[h];
        }
#pragma unroll
        for (int j = 0; j < 8; ++j) {
            int c = j * 32 + lane;
            acc[j] += alpha[c >> 6] * hh[(long)src * HH + c];
        }
    }
#pragma unroll
    for (int j = 0; j < 8; ++j) {
        int c = j * 32 + lane;
        out[(long)node * HH + c] = acc[j] + gat_b[c];
    }
}

// ---------------- gather propagation (TAG sym-norm / GraphConv sum) --------
template <bool NORM>
__global__ void __launch_bounds__(256)
propagate_kernel(const float* __restrict__ hin, const int* __restrict__ rowptr,
                 const int* __restrict__ csr_src, const float* __restrict__ dinv,
                 float* __restrict__ hout, int n) {
    int wid = blockIdx.x * (blockDim.x >> 5) + (threadIdx.x >> 5);
    if (wid >= n) return;
    int lane = threadIdx.x & 31;
    int start = rowptr[wid], end = rowptr[wid + 1];
    float dn = NORM ? dinv[wid] : 1.0f;
    float acc[8] = {0.f, 0.f, 0.f, 0.f, 0.f, 0.f, 0.f, 0.f};
    for (int i = start; i < end; ++i) {
        int src = csr_src[i];
        float w = NORM ? dn * dinv[src] : 1.0f;
#pragma unroll
        for (int j = 0; j < 8; ++j)
            acc[j] += w * hin[(long)src * HH + j * 32 + lane];
    }
#pragma unroll
    for (int j = 0; j < 8; ++j)
        hout[(long)wid * HH + j * 32 + lane] = acc[j];
}

// ---------------- graph_norm + relu (block per graph, thread per feature) --
__global__ void __launch_bounds__(256)
graphnorm_relu_kernel(const float* __restrict__ X, const float* __restrict__ w,
                      const float* __restrict__ b, const float* __restrict__ ms,
                      float* __restrict__ Y, int n, int g) {
    int gid = blockIdx.x;
    int f = threadIdx.x;
    long s = ((long)gid * n + g - 1) / g;        // batch[i] = (i*g)/n, sorted
    long e = ((long)(gid + 1) * n + g - 1) / g;
    float cnt = (float)(e - s);
    float sum = 0.f;
    for (long i = s; i < e; ++i) sum += X[i * HH + f];
    float mean = sum / cnt;
    float msf = ms[f];
    float var = 0.f;
    for (long i = s; i < e; ++i) {
        float o = X[i * HH + f] - msf * mean;
        var += o * o;
    }
    var /= cnt;
    float scale = rsqrtf(var + 1e-5f) * w[f];
    float bf = b[f];
    for (long i = s; i < e; ++i) {
        float o = X[i * HH + f] - msf * mean;
        Y[i * HH + f] = fmaxf(o * scale + bf, 0.f);
    }
}

// ---------------- global max/mean pool per graph ---------------------------
__global__ void __launch_bounds__(256)
pool_kernel(const float* __restrict__ X, float* __restrict__ xp, int n, int g) {
    int gid = blockIdx.x;
    int f = threadIdx.x;
    long s = ((long)gid * n + g - 1) / g;
    long e = ((long)(gid + 1) * n + g - 1) / g;
    float mx = -INFINITY, sum = 0.f;
    for (long i = s; i < e; ++i) {
        float v = X[i * HH + f];
        mx = fmaxf(mx, v);
        sum += v;
    }
    xp[(long)gid * (2 * HH) + f] = mx;
    xp[(long)gid * (2 * HH) + HH + f] = sum / (float)(e - s);
}

// ---------------- final linear (H->2) + log_softmax ------------------------
__global__ void cls2_kernel(const float* __restrict__ z, const float* __restrict__ w2,
                            const float* __restrict__ b2, float* __restrict__ out, int g) {
    int gid = blockIdx.x * blockDim.x + threadIdx.x;
    if (gid >= g) return;
    float z0 = b2[0], z1 = b2[1];
    const float* zr = z + (long)gid * HH;
    for (int f = 0; f < HH; ++f) {
        float v = zr[f];
        z0 += v * w2[f * 2];
        z1 += v * w2[f * 2 + 1];
    }
    float m = fmaxf(z0, z1);
    float lse = m + logf(expf(z0 - m) + expf(z1 - m));
    out[gid * 2] = z0 - lse;
    out[gid * 2 + 1] = z1 - lse;
}

extern "C" void kernel_launch(void* const* d_in, const int* in_sizes, int n_in,
                              void* d_out, int out_size, void* d_ws, size_t ws_size,
                              hipStream_t stream) {
    (void)n_in; (void)out_size; (void)ws_size;
    const float* x = (const float*)d_in[0];
    const int* edge_index = (const int*)d_in[1];
    const float* enc_w = (const float*)d_in[4];
    const float* enc_b = (const float*)d_in[5];
    const float* gat_w = (const float*)d_in[6];
    const float* att_src = (const float*)d_in[7];
    const float* att_dst = (const float*)d_in[8];
    const float* gat_b = (const float*)d_in[9];
    const float* n1_w = (const float*)d_in[10];
    const float* n1_b = (const float*)d_in[11];
    const float* n1_ms = (const float*)d_in[12];
    const float* tag_ws = (const float*)d_in[13];
    const float* tag_b = (const float*)d_in[14];
    const float* n2_w = (const float*)d_in[15];
    const float* n2_b = (const float*)d_in[16];
    const float* n2_ms = (const float*)d_in[17];
    const float* gc_w_rel = (const float*)d_in[18];
    const float* gc_b_rel = (const float*)d_in[19];
    const float* gc_w_root = (const float*)d_in[20];
    const float* cls_w1 = (const float*)d_in[21];
    const float* cls_b1 = (const float*)d_in[22];
    const float* cls_w2 = (const float*)d_in[23];
    const float* cls_b2 = (const float*)d_in[24];

    int N = in_sizes[0] / 7;
    int E = in_sizes[1] / 2;
    const int G = 1024;
    const int* row = edge_index;       // sources
    const int* col = edge_index + E;   // destinations

    // ---- workspace carve-out ----
    char* ws = (char*)d_ws;
    auto alloc = [&](size_t bytes) -> char* {
        char* p = ws;
        ws += (bytes + 255) & ~(size_t)255;
        return p;
    };
    float* bufA = (float*)alloc((size_t)N * HH * 4);
    float* bufB = (float*)alloc((size_t)N * HH * 4);
    float* bufC = (float*)alloc((size_t)N * HH * 4);
    float* asrc = (float*)alloc((size_t)N * NHEADS * 4);
    float* adst = (float*)alloc((size_t)N * NHEADS * 4);
    int* deg = (int*)alloc((size_t)N * 4);
    float* dinv = (float*)alloc((size_t)N * 4);
    int* rowptr = (int*)alloc((size_t)(N + 1) * 4);
    int* fillpos = (int*)alloc((size_t)N * 4);
    int* csr_src = (int*)alloc((size_t)E * 4);
    float* xp = (float*)alloc((size_t)G * 2 * HH * 4);
    float* zbuf = (float*)alloc((size_t)G * HH * 4);

    // ---- CSR build (by destination), in-degree, dinv ----
    hipMemsetAsync(deg, 0, (size_t)N * 4, stream);
    hipMemsetAsync(fillpos, 0, (size_t)N * 4, stream);
    deg_kernel<<<(E + 255) / 256, 256, 0, stream>>>(col, deg, E);
    scan_kernel<<<1, 256, 0, stream>>>(deg, rowptr, N, E);
    fill_csr_kernel<<<(E + 255) / 256, 256, 0, stream>>>(row, col, rowptr, fillpos, csr_src, E);
    dinv_kernel<<<(N + 255) / 256, 256, 0, stream>>>(deg, dinv, N);

    // ---- encoder ----
    encoder_kernel<<<(N * HH + 255) / 256, 256, 0, stream>>>(x, enc_w, enc_b, bufA, N);

    dim3 ggrid(HH / 64, N / 32);
    int wgrid = (N + 7) / 8;

    // ---- GATConv ----
    wmma_gemm_kernel<0, false, false><<<ggrid, 256, 0, stream>>>(bufA, gat_w, nullptr, bufB, N, HH, HH);
    att_score_kernel<<<(N * NHEADS + 255) / 256, 256, 0, stream>>>(bufB, att_src, att_dst, asrc, adst, N);
    gat_aggregate_kernel<<<wgrid, 256, 0, stream>>>(bufB, asrc, adst, rowptr, csr_src, gat_b, bufC, N);
    graphnorm_relu_kernel<<<G, HH, 0, stream>>>(bufC, n1_w, n1_b, n1_ms, bufA, N, G);

    // ---- TAGConv K=3: bufB = h@W0 + b; hk ping-pong A <-> C, accumulate into B ----
    wmma_gemm_kernel<0, false, true><<<ggrid, 256, 0, stream>>>(bufA, tag_ws + 0 * HH * HH, tag_b, bufB, N, HH, HH);
    propagate_kernel<true><<<wgrid, 256, 0, stream>>>(bufA, rowptr, csr_src, dinv, bufC, N);
    wmma_gemm_kernel<0, true, false><<<ggrid, 256, 0, stream>>>(bufC, tag_ws + 1 * HH * HH, nullptr, bufB, N, HH, HH);
    propagate_kernel<true><<<wgrid, 256, 0, stream>>>(bufC, rowptr, csr_src, dinv, bufA, N);
    wmma_gemm_kernel<0, true, false><<<ggrid, 256, 0, stream>>>(bufA, tag_ws + 2 * HH * HH, nullptr, bufB, N, HH, HH);
    propagate_kernel<true><<<wgrid, 256, 0, stream>>>(bufA, rowptr, csr_src, dinv, bufC, N);
    wmma_gemm_kernel<0, true, false><<<ggrid, 256, 0, stream>>>(bufC, tag_ws + 3 * HH * HH, nullptr, bufB, N, HH, HH);
    graphnorm_relu_kernel<<<G, HH, 0, stream>>>(bufB, n2_w, n2_b, n2_ms, bufA, N, G);

    // ---- GraphConv: C = relu(agg@W_rel + b_rel + h@W_root) ----
    propagate_kernel<false><<<wgrid, 256, 0, stream>>>(bufA, rowptr, csr_src, dinv, bufB, N);
    wmma_gemm_kernel<0, false, false><<<ggrid, 256, 0, stream>>>(bufA, gc_w_root, nullptr, bufC, N, HH, HH);
    wmma_gemm_kernel<1, true, true><<<ggrid, 256, 0, stream>>>(bufB, gc_w_rel, gc_b_rel, bufC, N, HH, HH);

    // ---- pool + classifier ----
    pool_kernel<<<G, HH, 0, stream>>>(bufC, xp, N, G);
    dim3 cgrid(HH / 64, G / 32);
    wmma_gemm_kernel<2, false, true><<<cgrid, 256, 0, stream>>>(xp, cls_w1, cls_b1, zbuf, G, 2 * HH, HH);
    cls2_kernel<<<(G + 255) / 256, 256, 0, stream>>>(zbuf, cls_w2, cls_b2, (float*)d_out, G);
}